// GAT_flatten_50835232916297
// MI455X (gfx1250) — compile-verified
//
#include <hip/hip_runtime.h>
#include <math.h>

#define IN_F   256
#define HID    32
#define OUTF   16
#define HEADS1 4
#define F1     128            // HEADS1*HID
#define NEG_SLOPE 0.2f

typedef float v2f __attribute__((ext_vector_type(2)));
typedef float v8f __attribute__((ext_vector_type(8)));

__device__ __forceinline__ float lrelu(float v) {
  return v > 0.f ? v : NEG_SLOPE * v;
}

// ---------------------------------------------------------------------------
// GEMM1: feat1[N,128] = X[N,256] @ W1[256,128]  (fp32 WMMA 16x16x4)
// grid.x = N/16 row tiles; 8 waves/block, wave w owns 16-col tile w.
// A layout (16x4 f32): lane&15 = M; lane>>4 selects K {0,1} vs {2,3}.
// B layout (4x16 f32): symmetric; C/D: VGPR j -> row j (lanes 0-15) / j+8.
// ---------------------------------------------------------------------------
__global__ void gemm1_wmma(const float* __restrict__ X,
                           const float* __restrict__ W,
                           float* __restrict__ F) {
  const int lane = threadIdx.x & 31;
  const int wave = threadIdx.x >> 5;          // 0..7 -> column tile
  const int row0 = blockIdx.x << 4;
  const int col0 = wave << 4;
  const int m    = lane & 15;                 // M for A, N for B/C
  const int kk   = (lane >> 4) << 1;          // 0 or 2
  const float* xrow = X + (size_t)(row0 + m) * IN_F;
  v8f acc = {};
  for (int k = 0; k < IN_F; k += 4) {
    v2f a, b;
    a.x = xrow[k + kk];
    a.y = xrow[k + kk + 1];
    b.x = W[(size_t)(k + kk)     * F1 + col0 + m];
    b.y = W[(size_t)(k + kk + 1) * F1 + col0 + m];
    acc = __builtin_amdgcn_wmma_f32_16x16x4_f32(false, a, false, b,
                                                (short)0, acc, false, false);
  }
  const int mo = (lane >> 4) << 3;
#pragma unroll
  for (int j = 0; j < 8; ++j)
    F[(size_t)(row0 + mo + j) * F1 + col0 + m] = acc[j];
}

// ---------------------------------------------------------------------------
// GEMM2: feat2[N,16] = H[N,128] @ W2[128,16]  (fp32 WMMA 16x16x4)
// ---------------------------------------------------------------------------
__global__ void gemm2_wmma(const float* __restrict__ Hh,
                           const float* __restrict__ W,
                           float* __restrict__ F, int n_tiles) {
  const int lane = threadIdx.x & 31;
  const int wave = threadIdx.x >> 5;
  const int tile = blockIdx.x * 8 + wave;     // wave-uniform -> EXEC stays full
  if (tile >= n_tiles) return;
  const int row0 = tile << 4;
  const int m    = lane & 15;
  const int kk   = (lane >> 4) << 1;
  const float* hrow = Hh + (size_t)(row0 + m) * F1;
  v8f acc = {};
  for (int k = 0; k < F1; k += 4) {
    v2f a, b;
    a.x = hrow[k + kk];
    a.y = hrow[k + kk + 1];
    b.x = W[(size_t)(k + kk)     * OUTF + m];
    b.y = W[(size_t)(k + kk + 1) * OUTF + m];
    acc = __builtin_amdgcn_wmma_f32_16x16x4_f32(false, a, false, b,
                                                (short)0, acc, false, false);
  }
  const int mo = (lane >> 4) << 3;
#pragma unroll
  for (int j = 0; j < 8; ++j)
    F[(size_t)(row0 + mo + j) * OUTF + m] = acc[j];
}

// ---------------------------------------------------------------------------
__global__ void fill_f32(float* __restrict__ p, float v, int n) {
  int i = blockIdx.x * blockDim.x + threadIdx.x;
  if (i < n) p[i] = v;
}

// el[i,h] = sum_d feat[i,h,d]*al[h,d];  er likewise
template<int NH, int DIM>
__global__ void attn_le(const float* __restrict__ feat,
                        const float* __restrict__ al, const float* __restrict__ ar,
                        float* __restrict__ el, float* __restrict__ er, int n) {
  int idx = blockIdx.x * blockDim.x + threadIdx.x;
  if (idx >= n * NH) return;
  int i = idx / NH, h = idx - i * NH;
  const float* f = feat + (size_t)i * (NH * DIM) + h * DIM;
  float sl = 0.f, sr = 0.f;
#pragma unroll 8
  for (int d = 0; d < DIM; ++d) {
    float v = f[d];
    sl += v * al[h * DIM + d];
    sr += v * ar[h * DIM + d];
  }
  el[idx] = sl;
  er[idx] = sr;
}

// ============================ Layer-1 edge ops (H=4, float4 per edge) ======
__global__ void edge_max4(const int* __restrict__ src, const int* __restrict__ dst,
                          const float4* __restrict__ el4, const float4* __restrict__ er4,
                          float* __restrict__ nmax, int E) {
  int e = blockIdx.x * blockDim.x + threadIdx.x;
  if (e >= E) return;
  int s = src[e], d = dst[e];
  float4 l = el4[s], r = er4[d];
  atomicMax(&nmax[d * 4 + 0], lrelu(l.x + r.x));
  atomicMax(&nmax[d * 4 + 1], lrelu(l.y + r.y));
  atomicMax(&nmax[d * 4 + 2], lrelu(l.z + r.z));
  atomicMax(&nmax[d * 4 + 3], lrelu(l.w + r.w));
}

__global__ void edge_exp4(const int* __restrict__ src, const int* __restrict__ dst,
                          const float4* __restrict__ el4, const float4* __restrict__ er4,
                          const float4* __restrict__ nmax4, float* __restrict__ nsum,
                          float4* __restrict__ ex4, int E) {
  int e = blockIdx.x * blockDim.x + threadIdx.x;
  if (e >= E) return;
  int s = src[e], d = dst[e];
  float4 l = el4[s], r = er4[d], mx = nmax4[d];
  float4 x;
  x.x = expf(lrelu(l.x + r.x) - mx.x);
  x.y = expf(lrelu(l.y + r.y) - mx.y);
  x.z = expf(lrelu(l.z + r.z) - mx.z);
  x.w = expf(lrelu(l.w + r.w) - mx.w);
  ex4[e] = x;
  atomicAdd(&nsum[d * 4 + 0], x.x);
  atomicAdd(&nsum[d * 4 + 1], x.y);
  atomicAdd(&nsum[d * 4 + 2], x.z);
  atomicAdd(&nsum[d * 4 + 3], x.w);
}

__global__ void edge_norm4(const int* __restrict__ dst,
                           const float4* __restrict__ nsum4,
                           float4* __restrict__ ex4, int E) {
  int e = blockIdx.x * blockDim.x + threadIdx.x;
  if (e >= E) return;
  float4 s = nsum4[dst[e]];
  float4 x = ex4[e];
  x.x /= s.x; x.y /= s.y; x.z /= s.z; x.w /= s.w;
  ex4[e] = x;
}

// one WAVE per edge: edge index is wave-uniform -> src/dst scalarize to SMEM;
// each lane gathers a float4 (global_load_b128) of feat1[src], scatters 4
// atomic adds into agg1[dst]. feat1/agg1 (25.6 MB each) are L2-resident.
__global__ void edge_agg1(const int* __restrict__ src, const int* __restrict__ dst,
                          const float4* __restrict__ feat4, const float* __restrict__ alpha,
                          float* __restrict__ agg, int E) {
  int wid = blockIdx.x * (blockDim.x >> 5) + (threadIdx.x >> 5);  // edge id, uniform
  if (wid >= E) return;
  int lane = threadIdx.x & 31;
  int s = src[wid], d = dst[wid];
  int h = lane >> 3;                                   // c = lane*4 -> head = lane/8
  float a = alpha[wid * 4 + h];
  float4 f = feat4[(size_t)s * 32 + lane];             // 32 float4 = 128 floats/row
  float* ap = agg + (size_t)d * F1 + lane * 4;
  atomicAdd(ap + 0, f.x * a);
  atomicAdd(ap + 1, f.y * a);
  atomicAdd(ap + 2, f.z * a);
  atomicAdd(ap + 3, f.w * a);
}

// ============================ Layer-2 edge ops (H=1) =======================
__global__ void edge_max1(const int* __restrict__ src, const int* __restrict__ dst,
                          const float* __restrict__ el, const float* __restrict__ er,
                          float* __restrict__ nmax, int E) {
  int e = blockIdx.x * blockDim.x + threadIdx.x;
  if (e >= E) return;
  atomicMax(&nmax[dst[e]], lrelu(el[src[e]] + er[dst[e]]));
}

__global__ void edge_exp1(const int* __restrict__ src, const int* __restrict__ dst,
                          const float* __restrict__ el, const float* __restrict__ er,
                          const float* __restrict__ nmax, float* __restrict__ nsum,
                          float* __restrict__ ex, int E) {
  int e = blockIdx.x * blockDim.x + threadIdx.x;
  if (e >= E) return;
  int d = dst[e];
  float x = expf(lrelu(el[src[e]] + er[d]) - nmax[d]);
  ex[e] = x;
  atomicAdd(&nsum[d], x);
}

__global__ void edge_norm1(const int* __restrict__ dst,
                           const float* __restrict__ nsum,
                           float* __restrict__ ex, int E) {
  int e = blockIdx.x * blockDim.x + threadIdx.x;
  if (e >= E) return;
  ex[e] = ex[e] / nsum[dst[e]];
}

// 4 threads per edge, each owns a float4 quarter of the 16-wide row
__global__ void edge_agg2(const int* __restrict__ src, const int* __restrict__ dst,
                          const float4* __restrict__ feat4, const float* __restrict__ alpha,
                          float* __restrict__ agg, int E) {
  int gid = blockIdx.x * blockDim.x + threadIdx.x;
  if (gid >= E * 4) return;
  int e = gid >> 2, q = gid & 3;
  float a = alpha[e];
  float4 f = feat4[(size_t)src[e] * 4 + q];
  float* ap = agg + (size_t)dst[e] * OUTF + q * 4;
  atomicAdd(ap + 0, f.x * a);
  atomicAdd(ap + 1, f.y * a);
  atomicAdd(ap + 2, f.z * a);
  atomicAdd(ap + 3, f.w * a);
}

// ---------------------------------------------------------------------------
__global__ void bias_elu4(float4* __restrict__ agg, const float4* __restrict__ b, int n4) {
  int idx = blockIdx.x * blockDim.x + threadIdx.x;
  if (idx >= n4) return;
  float4 v = agg[idx], bb = b[idx & (F1 / 4 - 1)];
  v.x += bb.x; v.y += bb.y; v.z += bb.z; v.w += bb.w;
  v.x = v.x > 0.f ? v.x : expm1f(v.x);
  v.y = v.y > 0.f ? v.y : expm1f(v.y);
  v.z = v.z > 0.f ? v.z : expm1f(v.z);
  v.w = v.w > 0.f ? v.w : expm1f(v.w);
  agg[idx] = v;
}

__global__ void bias_out4(const float4* __restrict__ agg, const float4* __restrict__ b,
                          float4* __restrict__ out, int n4) {
  int idx = blockIdx.x * blockDim.x + threadIdx.x;
  if (idx >= n4) return;
  float4 v = agg[idx], bb = b[idx & (OUTF / 4 - 1)];
  v.x += bb.x; v.y += bb.y; v.z += bb.z; v.w += bb.w;
  out[idx] = v;                                       // mean over 1 head = identity
}

// ---------------------------------------------------------------------------
extern "C" void kernel_launch(void* const* d_in, const int* in_sizes, int n_in,
                              void* d_out, int out_size, void* d_ws, size_t ws_size,
                              hipStream_t stream) {
  const float* x   = (const float*)d_in[0];
  const float* W1  = (const float*)d_in[1];
  const float* al1 = (const float*)d_in[2];
  const float* ar1 = (const float*)d_in[3];
  const float* b1  = (const float*)d_in[4];
  const float* W2  = (const float*)d_in[5];
  const float* al2 = (const float*)d_in[6];
  const float* ar2 = (const float*)d_in[7];
  const float* b2  = (const float*)d_in[8];
  const int*   src = (const int*)d_in[9];
  const int*   dst = (const int*)d_in[10];

  const int N = in_sizes[0] / IN_F;   // 50000 (divisible by 16)
  const int E = in_sizes[9];          // 800000

  char* ws = (char*)d_ws;
  size_t off = 0;
  auto take = [&](size_t bytes) -> float* {
    char* p = ws + off;
    off = (off + bytes + 255) & ~(size_t)255;
    return (float*)p;
  };
  float* feat1 = take((size_t)N * F1 * 4);        // 25.6 MB
  float* agg1  = take((size_t)N * F1 * 4);        // 25.6 MB (becomes h1 in place)
  float* el1   = take((size_t)N * HEADS1 * 4);
  float* er1   = take((size_t)N * HEADS1 * 4);
  float* nmax1 = take((size_t)N * HEADS1 * 4);
  float* nsum1 = take((size_t)N * HEADS1 * 4);
  float* ex1   = take((size_t)E * HEADS1 * 4);    // 12.8 MB
  float* feat2 = take((size_t)N * OUTF * 4);
  float* el2   = take((size_t)N * 4);
  float* er2   = take((size_t)N * 4);
  float* nmax2 = take((size_t)N * 4);
  float* nsum2 = take((size_t)N * 4);
  float* ex2   = take((size_t)E * 4);
  float* agg2  = take((size_t)N * OUTF * 4);

  const int B = 256;
  auto cdiv = [](long a, long b) { return (unsigned)((a + b - 1) / b); };
  const int ntile = N / 16;                       // 3125

  // ---- Layer 1 ----
  gemm1_wmma<<<ntile, B, 0, stream>>>(x, W1, feat1);
  attn_le<HEADS1, HID><<<cdiv((long)N * HEADS1, B), B, 0, stream>>>(feat1, al1, ar1, el1, er1, N);
  fill_f32<<<cdiv((long)N * HEADS1, B), B, 0, stream>>>(nmax1, -INFINITY, N * HEADS1);
  fill_f32<<<cdiv((long)N * HEADS1, B), B, 0, stream>>>(nsum1, 0.f, N * HEADS1);
  fill_f32<<<cdiv((long)N * F1, B), B, 0, stream>>>(agg1, 0.f, N * F1);
  edge_max4<<<cdiv(E, B), B, 0, stream>>>(src, dst, (const float4*)el1, (const float4*)er1, nmax1, E);
  edge_exp4<<<cdiv(E, B), B, 0, stream>>>(src, dst, (const float4*)el1, (const float4*)er1,
                                          (const float4*)nmax1, nsum1, (float4*)ex1, E);
  edge_norm4<<<cdiv(E, B), B, 0, stream>>>(dst, (const float4*)nsum1, (float4*)ex1, E);
  edge_agg1<<<cdiv((long)E * 32, B), B, 0, stream>>>(src, dst, (const float4*)feat1, ex1, agg1, E);
  bias_elu4<<<cdiv((long)N * F1 / 4, B), B, 0, stream>>>((float4*)agg1, (const float4*)b1, N * F1 / 4);

  // ---- Layer 2 ----
  gemm2_wmma<<<cdiv(ntile, 8), B, 0, stream>>>(agg1, W2, feat2, ntile);
  attn_le<1, OUTF><<<cdiv(N, B), B, 0, stream>>>(feat2, al2, ar2, el2, er2, N);
  fill_f32<<<cdiv(N, B), B, 0, stream>>>(nmax2, -INFINITY, N);
  fill_f32<<<cdiv(N, B), B, 0, stream>>>(nsum2, 0.f, N);
  fill_f32<<<cdiv((long)N * OUTF, B), B, 0, stream>>>(agg2, 0.f, N * OUTF);
  edge_max1<<<cdiv(E, B), B, 0, stream>>>(src, dst, el2, er2, nmax2, E);
  edge_exp1<<<cdiv(E, B), B, 0, stream>>>(src, dst, el2, er2, nmax2, nsum2, ex2, E);
  edge_norm1<<<cdiv(E, B), B, 0, stream>>>(dst, nsum2, ex2, E);
  edge_agg2<<<cdiv((long)E * 4, B), B, 0, stream>>>(src, dst, (const float4*)feat2, ex2, agg2, E);
  bias_out4<<<cdiv((long)N * OUTF / 4, B), B, 0, stream>>>((const float4*)agg2, (const float4*)b2,
                                                           (float4*)d_out, N * OUTF / 4);
}